// NeuralKB_37701222924630
// MI455X (gfx1250) — compile-verified
//
#include <hip/hip_runtime.h>
#include <hip/hip_bf16.h>

// Problem dims fixed by the reference.
constexpr int BB = 8;      // batch
constexpr int NN = 4000;   // entities
constexpr int FF = 4000;   // facts
constexpr int EE = 100;    // embedding dim
constexpr int KSTEPS = EE / 4;  // 25 WMMA f32 16x16x4 K-steps

typedef float v2f __attribute__((ext_vector_type(2)));
typedef float v8f __attribute__((ext_vector_type(8)));

// ---------------------------------------------------------------------------
// Precompute 1: cc_sp[f][b] = ||fact_f||^2 - 2*(rel_b.frel_f + arg1_b.farg1_f)
//               cc_po[f][b] = ||fact_f||^2 - 2*(rel_b.frel_f + arg2_b.farg2_f)
// ---------------------------------------------------------------------------
__global__ void precomp_cc_kernel(const float* __restrict__ rel,
                                  const float* __restrict__ arg1,
                                  const float* __restrict__ arg2,
                                  const float* __restrict__ frel,
                                  const float* __restrict__ farg1,
                                  const float* __restrict__ farg2,
                                  float* __restrict__ cc_sp,
                                  float* __restrict__ cc_po) {
  int f = blockIdx.x * blockDim.x + threadIdx.x;
  if (f >= FF) return;
  float f2 = 0.0f;
  float dr[BB], d1[BB], d2[BB];
#pragma unroll
  for (int b = 0; b < BB; ++b) { dr[b] = 0.0f; d1[b] = 0.0f; d2[b] = 0.0f; }
  const float* fr = frel  + (size_t)f * EE;
  const float* f1 = farg1 + (size_t)f * EE;
  const float* fa = farg2 + (size_t)f * EE;
  for (int e = 0; e < EE; ++e) {
    float vr = fr[e], v1 = f1[e], va = fa[e];
    f2 = fmaf(vr, vr, f2); f2 = fmaf(v1, v1, f2); f2 = fmaf(va, va, f2);
#pragma unroll
    for (int b = 0; b < BB; ++b) {
      dr[b] = fmaf(rel[b * EE + e], vr, dr[b]);
      d1[b] = fmaf(arg1[b * EE + e], v1, d1[b]);
      d2[b] = fmaf(arg2[b * EE + e], va, d2[b]);
    }
  }
#pragma unroll
  for (int b = 0; b < BB; ++b) {
    cc_sp[(size_t)f * 8 + b] = f2 - 2.0f * (dr[b] + d1[b]);
    cc_po[(size_t)f * 8 + b] = f2 - 2.0f * (dr[b] + d2[b]);
  }
}

// ---------------------------------------------------------------------------
// Precompute 2: e2[n] = ||ent_n||^2
// ---------------------------------------------------------------------------
__global__ void precomp_e2_kernel(const float* __restrict__ ent,
                                  float* __restrict__ e2) {
  int n = blockIdx.x * blockDim.x + threadIdx.x;
  if (n >= NN) return;
  const float* row = ent + (size_t)n * EE;
  float s = 0.0f;
  for (int e = 0; e < EE; ++e) s = fmaf(row[e], row[e], s);
  e2[n] = s;
}

// ---------------------------------------------------------------------------
// Precompute 3: q2[0..7] = ||rel_b||^2+||arg1_b||^2 ; q2[8..15] = ||rel_b||^2+||arg2_b||^2
// ---------------------------------------------------------------------------
__global__ void precomp_q2_kernel(const float* __restrict__ rel,
                                  const float* __restrict__ arg1,
                                  const float* __restrict__ arg2,
                                  float* __restrict__ q2) {
  int t = threadIdx.x;
  if (t >= 2 * BB) return;
  int b = t & (BB - 1);
  const float* other = (t < BB) ? (arg1 + (size_t)b * EE) : (arg2 + (size_t)b * EE);
  const float* r = rel + (size_t)b * EE;
  float s = 0.0f;
  for (int e = 0; e < EE; ++e) {
    s = fmaf(r[e], r[e], s);
    s = fmaf(other[e], other[e], s);
  }
  q2[t] = s;
}

// ---------------------------------------------------------------------------
// Main fused kernel: block = 4 waves, one 16-entity tile per block.
// M = fact rows (A operand), N = entity cols (B operand, register-resident).
// Each wave strides over fact 16-row tiles, runs 25 chained
// v_wmma_f32_16x16x4_f32 per gram (ent.farg1 and ent.farg2, sharing B),
// and folds the min-over-f reduction into the epilogue.
//
// WMMA f32 16x16x4 layout assumptions (per CDNA5 ISA):
//   A (16x4): lane L<16 holds {A[L][0],A[L][1]}, lane L>=16 holds {A[L-16][2],A[L-16][3]}
//   B (4x16): lane L<16 holds {B[0][L],B[1][L]},  lane L>=16 holds {B[2][L-16],B[3][L-16]}
//   C/D (16x16): lane L, VGPR v -> row M = v + 8*(L>=16), col N = L%16
// ---------------------------------------------------------------------------
__global__ void __launch_bounds__(128)
neuralkb_score_kernel(const float* __restrict__ ent,    // [N][E]
                      const float* __restrict__ farg1,  // [F][E]
                      const float* __restrict__ farg2,  // [F][E]
                      const float* __restrict__ cc_sp,  // [F][8]
                      const float* __restrict__ cc_po,  // [F][8]
                      const float* __restrict__ e2,     // [N]
                      const float* __restrict__ q2,     // [16]
                      float* __restrict__ out)          // [2][B][N]
{
  const int lane = threadIdx.x & 31;
  const int wave = threadIdx.x >> 5;   // 0..3
  const int half = lane >> 4;          // 0 or 1
  const int lmod = lane & 15;
  const int n_base = blockIdx.x * 16;

  // --- B operand: entity tile, held in 50 VGPRs for the whole kernel ---
  v2f breg[KSTEPS];
  const float* entrow = ent + (size_t)(n_base + lmod) * EE + 2 * half;
#pragma unroll
  for (int j = 0; j < KSTEPS; ++j)
    breg[j] = *reinterpret_cast<const v2f*>(entrow + 4 * j);

  float msp[BB], mpo[BB];
#pragma unroll
  for (int b = 0; b < BB; ++b) { msp[b] = 3.402823466e38f; mpo[b] = 3.402823466e38f; }

  for (int ft = wave; ft < FF / 16; ft += 4) {
    const int f_base = ft * 16;
    v8f acc1 = {};  // ent . farg1  (used by score_po)
    v8f acc2 = {};  // ent . farg2  (used by score_sp)
    const float* a1row = farg1 + (size_t)(f_base + lmod) * EE + 2 * half;
    const float* a2row = farg2 + (size_t)(f_base + lmod) * EE + 2 * half;
#pragma unroll
    for (int j = 0; j < KSTEPS; ++j) {
      v2f a1 = *reinterpret_cast<const v2f*>(a1row + 4 * j);
      v2f a2 = *reinterpret_cast<const v2f*>(a2row + 4 * j);
      acc1 = __builtin_amdgcn_wmma_f32_16x16x4_f32(
          false, a1, false, breg[j], (short)0, acc1, false, false);
      acc2 = __builtin_amdgcn_wmma_f32_16x16x4_f32(
          false, a2, false, breg[j], (short)0, acc2, false, false);
    }
    // Epilogue: this lane's D rows are fact rows f_base + v + 8*half.
#pragma unroll
    for (int v = 0; v < 8; ++v) {
      const int fr = f_base + v + 8 * half;
      const float* cs = cc_sp + (size_t)fr * 8;
      const float* cp = cc_po + (size_t)fr * 8;
      const float t_sp = -2.0f * acc2[v];
      const float t_po = -2.0f * acc1[v];
#pragma unroll
      for (int b = 0; b < BB; ++b) {
        msp[b] = fminf(msp[b], cs[b] + t_sp);
        mpo[b] = fminf(mpo[b], cp[b] + t_po);
      }
    }
  }

  // Merge the two lane-halves (rows 0-7 vs 8-15 of each fact tile).
#pragma unroll
  for (int b = 0; b < BB; ++b) {
    msp[b] = fminf(msp[b], __shfl_xor(msp[b], 16, 32));
    mpo[b] = fminf(mpo[b], __shfl_xor(mpo[b], 16, 32));
  }

  // Cross-wave merge via LDS, then finalize.
  __shared__ float lds_sp[4][16][BB];
  __shared__ float lds_po[4][16][BB];
  if (lane < 16) {
#pragma unroll
    for (int b = 0; b < BB; ++b) {
      lds_sp[wave][lane][b] = msp[b];
      lds_po[wave][lane][b] = mpo[b];
    }
  }
  __syncthreads();
  if (wave == 0 && lane < 16) {
    const int n = n_base + lane;
    const float en2 = e2[n];
#pragma unroll
    for (int b = 0; b < BB; ++b) {
      float a = fminf(fminf(lds_sp[0][lane][b], lds_sp[1][lane][b]),
                      fminf(lds_sp[2][lane][b], lds_sp[3][lane][b]));
      float p = fminf(fminf(lds_po[0][lane][b], lds_po[1][lane][b]),
                      fminf(lds_po[2][lane][b], lds_po[3][lane][b]));
      float d2s = fmaxf(q2[b] + en2 + a, 0.0f);
      float d2p = fmaxf(q2[BB + b] + en2 + p, 0.0f);
      out[(size_t)b * NN + n]                     = expf(-0.5f * d2s);
      out[(size_t)(BB + b) * NN + n]              = expf(-0.5f * d2p);
    }
  }
}

// ---------------------------------------------------------------------------
// Launcher
// ---------------------------------------------------------------------------
extern "C" void kernel_launch(void* const* d_in, const int* in_sizes, int n_in,
                              void* d_out, int out_size, void* d_ws, size_t ws_size,
                              hipStream_t stream) {
  const float* rel   = (const float*)d_in[0];  // [B,E]
  const float* arg1  = (const float*)d_in[1];  // [B,E]
  const float* arg2  = (const float*)d_in[2];  // [B,E]
  const float* frel  = (const float*)d_in[3];  // [F,E]
  const float* farg1 = (const float*)d_in[4];  // [F,E]
  const float* farg2 = (const float*)d_in[5];  // [F,E]
  const float* ent   = (const float*)d_in[6];  // [N,E]
  float* out = (float*)d_out;                  // [2,B,N]

  // Workspace carve-up (floats): cc_sp[F*8] | cc_po[F*8] | e2[N] | q2[16]
  float* cc_sp = (float*)d_ws;
  float* cc_po = cc_sp + (size_t)FF * 8;
  float* e2    = cc_po + (size_t)FF * 8;
  float* q2    = e2 + NN;

  precomp_cc_kernel<<<(FF + 63) / 64, 64, 0, stream>>>(
      rel, arg1, arg2, frel, farg1, farg2, cc_sp, cc_po);
  precomp_e2_kernel<<<(NN + 255) / 256, 256, 0, stream>>>(ent, e2);
  precomp_q2_kernel<<<1, 32, 0, stream>>>(rel, arg1, arg2, q2);
  neuralkb_score_kernel<<<NN / 16, 128, 0, stream>>>(
      ent, farg1, farg2, cc_sp, cc_po, e2, q2, out);
}